// GlimpseNetwork_16758962389780
// MI455X (gfx1250) — compile-verified
//
#include <hip/hip_runtime.h>

// ---------------------------------------------------------------------------
// Glimpse network forward (MI455X / gfx1250, wave32, fp32 WMMA 16x16x4)
//   g_out = relu(glimpse(x, l_prev) @ W1 + b1)                    (256 x 128)
//   out   = relu(g_out @ W3 + relu(l_prev@W2+b2) @ W4 + b3 + b4)  (256 x 256)
// ---------------------------------------------------------------------------

typedef float v2f __attribute__((ext_vector_type(2)));
typedef float v8f __attribute__((ext_vector_type(8)));

#define GSZ   32       // glimpse size
#define CH    3
#define IMH   224
#define IMW   224
#define HG    128
#define HL    128
#define HOUT  256      // HG + HL
#define NB    256      // batch

__device__ __forceinline__ int clampi(int v, int lo, int hi) {
  v = v < lo ? lo : v;
  return v > hi ? hi : v;
}

// start = trunc(0.5*((l+1)*224)), then dynamic_slice clamp to [0, 224]
__device__ __forceinline__ int start_coord(float v) {
  return clampi((int)(0.5f * ((v + 1.0f) * (float)IMH)), 0, IMH);
}

// ---- Kernel 1: g_out = relu(glimpse @ W1 + b1), one wave per 16x16 tile ----
// Patch extraction fused into the A-operand loads, branch-free:
// clamped-address load + select-0 (no EXEC divergence around the WMMAs).
__global__ __launch_bounds__(32)
void glimpse_gemm1(const float* __restrict__ x, const float* __restrict__ l_prev,
                   const float* __restrict__ W1, const float* __restrict__ b1,
                   float* __restrict__ g_out) {
  const int lane = threadIdx.x;        // 0..31
  const int mn   = lane & 15;          // A-row in tile == B-col in tile
  const int half = lane >> 4;          // K-half select (K = {0,1} or {2,3})
  const int m0   = blockIdx.y * 16;
  const int n0   = blockIdx.x * 16;
  const int row  = m0 + mn;            // batch index for this lane's A row
  const int col  = n0 + mn;            // output column for this lane's B col

  const int sx = start_coord(l_prev[row * 2 + 0]);   // l[:,0] = x (cols)
  const int sy = start_coord(l_prev[row * 2 + 1]);   // l[:,1] = y (rows)

  // Per-lane bases, hoisted out of all loops.
  const float* __restrict__ xrow = x + (size_t)row * CH * IMH * IMW;
  const float* __restrict__ wb   = W1 + col + 2 * half * HG;  // b.x base; b.y = +HG
  const int wofs = sx + 2 * half - (GSZ / 2);                 // lane's w offset

  v8f acc = {};
  for (int c = 0; c < CH; ++c) {
    for (int gy = 0; gy < GSZ; ++gy) {
      const int h = sy + gy - (GSZ / 2);
      const bool hok = (unsigned)h < (unsigned)IMH;
      // Always-valid row pointer (row 0 when h is out of range).
      const float* __restrict__ rp = xrow + ((size_t)c * IMH + (hok ? h : 0)) * IMW;
      const float* __restrict__ wp = wb + (size_t)((c * GSZ + gy) * GSZ) * HG;
#pragma unroll
      for (int gxb = 0; gxb < GSZ; gxb += 4) {
        const int w0 = wofs + gxb;          // k = c*1024 + gy*32 + gxb + 2*half
        const int w1 = w0 + 1;
        // Clamped loads (always in-bounds), then select 0 for OOB elements.
        const float v0 = rp[clampi(w0, 0, IMW - 1)];
        const float v1 = rp[clampi(w1, 0, IMW - 1)];
        v2f a, b;
        a.x = (hok && (unsigned)w0 < (unsigned)IMW) ? v0 : 0.0f;
        a.y = (hok && (unsigned)w1 < (unsigned)IMW) ? v1 : 0.0f;
        b.x = wp[(size_t)gxb * HG];
        b.y = wp[(size_t)gxb * HG + HG];
        acc = __builtin_amdgcn_wmma_f32_16x16x4_f32(false, a, false, b,
                                                    (short)0, acc, false, false);
      }
    }
  }

  const float bias = b1[col];
#pragma unroll
  for (int r = 0; r < 8; ++r) {
    const float v = acc[r] + bias;
    g_out[(m0 + r + 8 * half) * HG + col] = v > 0.0f ? v : 0.0f;
  }
}

// ---- Kernel 2: out = relu(g_out@W3 + relu(l_prev@W2+b2)@W4 + b3 + b4) -----
__global__ __launch_bounds__(32)
void glimpse_gemm2(const float* __restrict__ g_out, const float* __restrict__ l_prev,
                   const float* __restrict__ W2, const float* __restrict__ b2,
                   const float* __restrict__ W3, const float* __restrict__ b3,
                   const float* __restrict__ W4, const float* __restrict__ b4,
                   float* __restrict__ out) {
  const int lane = threadIdx.x;
  const int mn   = lane & 15;
  const int half = lane >> 4;
  const int m0   = blockIdx.y * 16;
  const int n0   = blockIdx.x * 16;
  const int row  = m0 + mn;
  const int col  = n0 + mn;

  const float lx = l_prev[row * 2 + 0];
  const float ly = l_prev[row * 2 + 1];

  // Hoisted per-lane bases: everything below uses immediate-style offsets.
  const float* __restrict__ ga  = g_out + row * HG + 2 * half;      // a.x; a.y = +1
  const float* __restrict__ w3p = W3 + col + 2 * half * HOUT;       // b.x; b.y = +HOUT
  const float* __restrict__ w4p = W4 + col + 2 * half * HOUT;
  const float* __restrict__ w2a = W2 + 2 * half;                    // row 0 of W2
  const float* __restrict__ w2b = W2 + HL + 2 * half;               // row 1 of W2
  const float* __restrict__ b2p = b2 + 2 * half;

  v8f acc = {};

  // what = g_out @ W3   (K = 128)
#pragma unroll 8
  for (int kk = 0; kk < HG; kk += 4) {
    v2f a, b;
    a.x = ga[kk];
    a.y = ga[kk + 1];
    b.x = w3p[(size_t)kk * HOUT];
    b.y = w3p[(size_t)kk * HOUT + HOUT];
    acc = __builtin_amdgcn_wmma_f32_16x16x4_f32(false, a, false, b,
                                                (short)0, acc, false, false);
  }

  // where = l_out @ W4, with l_out = relu(l_prev@W2 + b2) built on the fly.
#pragma unroll 8
  for (int kk = 0; kk < HL; kk += 4) {
    v2f a, b;
    const float t0 = fmaf(lx, w2a[kk],     fmaf(ly, w2b[kk],     b2p[kk]));
    const float t1 = fmaf(lx, w2a[kk + 1], fmaf(ly, w2b[kk + 1], b2p[kk + 1]));
    a.x = t0 > 0.0f ? t0 : 0.0f;
    a.y = t1 > 0.0f ? t1 : 0.0f;
    b.x = w4p[(size_t)kk * HOUT];
    b.y = w4p[(size_t)kk * HOUT + HOUT];
    acc = __builtin_amdgcn_wmma_f32_16x16x4_f32(false, a, false, b,
                                                (short)0, acc, false, false);
  }

  const float bias = b3[col] + b4[col];
#pragma unroll
  for (int r = 0; r < 8; ++r) {
    const float v = acc[r] + bias;
    out[(m0 + r + 8 * half) * HOUT + col] = v > 0.0f ? v : 0.0f;
  }
}

extern "C" void kernel_launch(void* const* d_in, const int* in_sizes, int n_in,
                              void* d_out, int out_size, void* d_ws, size_t ws_size,
                              hipStream_t stream) {
  const float* x      = (const float*)d_in[0];
  const float* l_prev = (const float*)d_in[1];
  const float* W1     = (const float*)d_in[2];
  const float* b1     = (const float*)d_in[3];
  const float* W2     = (const float*)d_in[4];
  const float* b2     = (const float*)d_in[5];
  const float* W3     = (const float*)d_in[6];
  const float* b3     = (const float*)d_in[7];
  const float* W4     = (const float*)d_in[8];
  const float* b4     = (const float*)d_in[9];

  float* out   = (float*)d_out;
  float* g_out = (float*)d_ws;                 // 256*128 floats = 128 KB scratch

  dim3 grid1(HG / 16, NB / 16);                // (8, 16) tiles, 128 waves
  glimpse_gemm1<<<grid1, dim3(32), 0, stream>>>(x, l_prev, W1, b1, g_out);

  dim3 grid2(HOUT / 16, NB / 16);              // (16, 16) tiles, 256 waves
  glimpse_gemm2<<<grid2, dim3(32), 0, stream>>>(g_out, l_prev, W2, b2,
                                                W3, b3, W4, b4, out);
}